// ScoringLayer_76459007804058
// MI455X (gfx1250) — compile-verified
//
#include <hip/hip_runtime.h>
#include <math.h>

typedef __attribute__((ext_vector_type(2))) float v2f;
typedef __attribute__((ext_vector_type(8))) float v8f;

#define BROWS 131072
#define NCOL  1280
// ws layout (float offsets)
#define WS_S1 0
#define WS_S0 1280
#define WS_R  2560
#define WS_V1 3840
#define WS_V0 5120
#define WS_W1 6400
#define WS_C1 6401
#define WS_C0 6402
#define WS_Y  6403   // 131072 dense y values

// ---------------- pass 0: zero the accumulators ----------------
__global__ void zero_ws_kernel(float* ws) {
    int i = blockIdx.x * blockDim.x + threadIdx.x;
    if (i < 2560) ws[i] = 0.0f;         // S1, S0
    if (i < 3)    ws[WS_W1 + i] = 0.0f; // w1, C1, C0
}

// ---------------- pass 1a: densify y, reduce w1 ----------------
__global__ void sum_y_kernel(const float* __restrict__ x0, float* __restrict__ ws) {
    __shared__ float red[256];
    int idx = blockIdx.x * 256 + threadIdx.x;
    float y = 0.0f;
    if (idx < BROWS) {
        y = x0[(size_t)idx * 65 + 64];
        ws[WS_Y + idx] = y;
    }
    red[threadIdx.x] = y;
    __syncthreads();
    for (int s = 128; s > 0; s >>= 1) {
        if (threadIdx.x < s) red[threadIdx.x] += red[threadIdx.x + s];
        __syncthreads();
    }
    if (threadIdx.x == 0) atomicAdd(&ws[WS_W1], red[0]);
}

// ---------------- pass 1b: class-conditional column sums ----------------
__global__ void colsum_kernel(const float* __restrict__ L1,
                              const float* __restrict__ L2,
                              const float* __restrict__ L3,
                              float* __restrict__ ws) {
    const int layer = blockIdx.y;
    const float* L = (layer == 0) ? L1 : (layer == 1) ? L2 : L3;
    const int d       = (layer == 2) ? 256 : 512;
    const int colbase = (layer == 0) ? 0 : (layer == 1) ? 512 : 1024;
    const int ng = d >> 8;                 // column groups of 256: 2 or 1
    const int r0 = blockIdx.x * 512;
    const float* __restrict__ yv = ws + WS_Y;

    float s[2]  = {0.0f, 0.0f};
    float s1[2] = {0.0f, 0.0f};
    for (int r = r0; r < r0 + 512; ++r) {
        float y = yv[r];
        const float* rowp = L + (size_t)r * d;
        #pragma unroll
        for (int g = 0; g < 2; ++g) {
            if (g < ng) {
                float v = __builtin_nontemporal_load(rowp + threadIdx.x + (g << 8));
                s[g]  += v;
                s1[g] += y * v;
            }
        }
    }
    for (int g = 0; g < ng; ++g) {
        int c = colbase + threadIdx.x + (g << 8);
        atomicAdd(&ws[WS_S1 + c], s1[g]);
        atomicAdd(&ws[WS_S0 + c], s[g] - s1[g]);
    }
}

// ---------------- pass 1c: per-column scale vectors + constants ----------------
__global__ void prep_kernel(float* __restrict__ ws) {
    float w1sum = ws[WS_W1];
    float w1 = w1sum + 1e-6f;
    float w0 = ((float)BROWS - w1sum) + 1e-6f;
    float c1 = 0.0f, c0 = 0.0f;
    for (int j = threadIdx.x; j < NCOL; j += 256) {
        int li, d;
        if (j < 512)       { li = 1; d = 512; }
        else if (j < 1024) { li = 2; d = 512; }
        else               { li = 3; d = 256; }
        float wgt = 0.1f * (float)li / (6.0f * (float)d);  // alpha * i / (layers_norm * d)
        float xb1 = ws[WS_S1 + j] / w1;
        float xb0 = ws[WS_S0 + j] / w0;
        float dlt = xb1 - xb0;
        float msd = dlt * dlt + 0.01f;
        float r = wgt / msd;
        ws[WS_R  + j] = r;
        ws[WS_V1 + j] = r * xb1;
        ws[WS_V0 + j] = r * xb0;
        c1 += r * xb1 * xb1;
        c0 += r * xb0 * xb0;
    }
    __shared__ float ra[256], rb[256];
    ra[threadIdx.x] = c1; rb[threadIdx.x] = c0;
    __syncthreads();
    for (int s = 128; s > 0; s >>= 1) {
        if (threadIdx.x < s) { ra[threadIdx.x] += ra[threadIdx.x + s];
                               rb[threadIdx.x] += rb[threadIdx.x + s]; }
        __syncthreads();
    }
    if (threadIdx.x == 0) { ws[WS_C1] = ra[0]; ws[WS_C0] = rb[0]; }
}

// ---------------- pass 2: WMMA row reductions + final score ----------------
// Each wave owns a 32-row tile: two stacked 16xK A-fragments that SHARE one
// (bD,bQ) B-fragment pair per K-chunk, halving B-load overhead per A-byte.
// Per K-chunk of 4 columns:
//   accD{t} += A{t}(16x4 of L)    x B(4x16; col0=r*xbar1, col1=r*xbar0, rest garbage)
//   accQ{t} += A{t}^2(16x4 of L2) x B(4x16; col0=r, rest garbage)
// Garbage columns are never read back, so B needs no masking (uniform flow).
// A layout (f32 16x4): lanes 0-15 = M, VGPR0 = K{0|2}, VGPR1 = K{1|3} by half.
// B layout (f32 4x16): lanes = N,      VGPR0 = K{0|2}, VGPR1 = K{1|3} by half.
__global__ void __launch_bounds__(256)
score_wmma_kernel(const float* __restrict__ L1, const float* __restrict__ L2,
                  const float* __restrict__ L3, const float* __restrict__ ypred,
                  const float* __restrict__ ws, float* __restrict__ out) {
    const int lane = threadIdx.x & 31;
    const int wave = threadIdx.x >> 5;
    const int row0 = (blockIdx.x * 8 + wave) * 32;
    const int half = lane >> 4;      // 0: K=0,1   1: K=2,3
    const int n    = lane & 15;      // A: M index / B: N index
    const int rowA = row0 + n;       // tile 0 rows
    const int rowB = rowA + 16;      // tile 1 rows

    const float* __restrict__ rr = ws + WS_R;
    const float* __restrict__ v1 = ws + WS_V1;
    const float* __restrict__ v0 = ws + WS_V0;

    // per-lane B source, fixed for the whole kernel (lanes n>=2: value irrelevant)
    const float* bsrcD = (n == 1) ? v0 : v1;

    v8f accD0 = {}, accQ0 = {};   // rows row0..row0+15
    v8f accD1 = {}, accQ1 = {};   // rows row0+16..row0+31

    const float* Ls[3] = {L1, L2, L3};
    const int    ds[3] = {512, 512, 256};
    int colbase = 0;
    #pragma unroll
    for (int l = 0; l < 3; ++l) {
        const int d = ds[l];
        const float* ap0 = Ls[l] + (size_t)rowA * d + (half << 1);
        const float* ap1 = Ls[l] + (size_t)rowB * d + (half << 1);
        const float* bDp = bsrcD + colbase + (half << 1);
        const float* bQp = rr    + colbase + (half << 1);
        #pragma unroll 4
        for (int kb = 0; kb < d; kb += 4) {
            v2f a0  = __builtin_nontemporal_load((const v2f*)(ap0 + kb));
            v2f a1  = __builtin_nontemporal_load((const v2f*)(ap1 + kb));
            v2f bD  = *(const v2f*)(bDp + kb);
            v2f bQ  = *(const v2f*)(bQp + kb);
            v2f q0  = a0 * a0;
            v2f q1  = a1 * a1;
            accD0 = __builtin_amdgcn_wmma_f32_16x16x4_f32(
                false, a0, false, bD, (short)0, accD0, false, false);
            accD1 = __builtin_amdgcn_wmma_f32_16x16x4_f32(
                false, a1, false, bD, (short)0, accD1, false, false);
            accQ0 = __builtin_amdgcn_wmma_f32_16x16x4_f32(
                false, q0, false, bQ, (short)0, accQ0, false, false);
            accQ1 = __builtin_amdgcn_wmma_f32_16x16x4_f32(
                false, q1, false, bQ, (short)0, accQ1, false, false);
        }
        colbase += d;
    }

    // D/C layout: VGPR p = row M=p (lanes 0-15, N=lane) and M=p+8 (lanes 16-31).
    // Gather so lane m (0..15) holds results for rows base+m; then finalize.
    const float C1v = ws[WS_C1];
    const float C0v = ws[WS_C0];
    #pragma unroll
    for (int t = 0; t < 2; ++t) {
        const v8f& aD = t ? accD1 : accD0;
        const v8f& aQ = t ? accQ1 : accQ0;
        float myD1 = 0.0f, myD0 = 0.0f, myQ = 0.0f;
        #pragma unroll
        for (int p = 0; p < 8; ++p) {
            float d1lo = __shfl(aD[p], 0, 32);
            float d0lo = __shfl(aD[p], 1, 32);
            float qlo  = __shfl(aQ[p], 0, 32);
            float d1hi = __shfl(aD[p], 16, 32);
            float d0hi = __shfl(aD[p], 17, 32);
            float qhi  = __shfl(aQ[p], 16, 32);
            if (lane == p)     { myD1 = d1lo; myD0 = d0lo; myQ = qlo; }
            if (lane == p + 8) { myD1 = d1hi; myD0 = d0hi; myQ = qhi; }
        }
        if (lane < 16) {
            int row = row0 + t * 16 + lane;
            float y  = ws[WS_Y + row];
            float yp = fminf(fmaxf(ypred[row], 1e-6f), 1.0f - 1e-6f);
            float loss = -(y * __logf(yp) + (1.0f - y) * __logf(1.0f - yp));
            float C  = (y > 0.5f) ? C1v : C0v;
            float Dm = (y > 0.5f) ? myD1 : myD0;
            float v  = loss + myQ - 2.0f * Dm + C;
            out[row] = sqrtf(fmaxf(v, 0.0f));
        }
    }
}

extern "C" void kernel_launch(void* const* d_in, const int* in_sizes, int n_in,
                              void* d_out, int out_size, void* d_ws, size_t ws_size,
                              hipStream_t stream) {
    const float* x0 = (const float*)d_in[0];
    const float* x1 = (const float*)d_in[1];
    const float* x2 = (const float*)d_in[2];
    const float* x3 = (const float*)d_in[3];
    const float* yp = (const float*)d_in[4];
    float* out = (float*)d_out;
    float* ws  = (float*)d_ws;

    zero_ws_kernel<<<10, 256, 0, stream>>>(ws);                    // 2560 threads
    sum_y_kernel<<<BROWS / 256, 256, 0, stream>>>(x0, ws);         // 512 blocks
    dim3 gcs(256, 3);
    colsum_kernel<<<gcs, 256, 0, stream>>>(x1, x2, x3, ws);        // 512-row blocks
    prep_kernel<<<1, 256, 0, stream>>>(ws);
    score_wmma_kernel<<<BROWS / 256, 256, 0, stream>>>(x1, x2, x3, yp, ws, out); // 512 blocks
}